// QMeasureDensity_30897994728226
// MI455X (gfx1250) — compile-verified
//
#include <hip/hip_runtime.h>
#include <hip/hip_bf16.h>

typedef float v2f __attribute__((ext_vector_type(2)));
typedef float v8f __attribute__((ext_vector_type(8)));

#define B_ROWS 16384
#define DDIM   2048
#define MB     32            // rows of X per workgroup
#define NWAVES 8             // 256 threads, wave32
#define NT     4             // concurrent N-tiles per wave
#define MT     2             // concurrent M-tiles per wave
#define NPASS  (DDIM / (16 * NWAVES * NT))   // 2048 / 512 = 4 N passes

// Fused kernel: quad[b] = x_b^T rho x_b via WMMA f32 GEMM tiles (xr never
// stored), norm2[b] = ||x_b||^2, out[b] = norm2[b] * quad[b].
__global__ __launch_bounds__(256) void QMeasureDensity_wmma_kernel(
    const float* __restrict__ X,   // (B_ROWS, DDIM)
    const float* __restrict__ R,   // (DDIM, DDIM) row-major
    float* __restrict__ out)       // (B_ROWS)
{
    __shared__ float s_norm2[MB];
    __shared__ float s_quad[NWAVES][MB];

    const int m0   = blockIdx.x * MB;
    const int tid  = threadIdx.x;
    const int wave = tid >> 5;
    const int lane = tid & 31;
    const int hi   = lane >> 4;   // which 16-lane half
    const int l15  = lane & 15;

    // ---------------- norm2: ||x_b||^2 (4 rows per wave, 8 lanes per row) ---
    {
        const int r  = (wave << 2) + (lane >> 3);   // 0..31 row within block
        const int c0 = lane & 7;
        const float* xp = X + (size_t)(m0 + r) * DDIM;
        float s = 0.f;
        #pragma unroll 8
        for (int d = c0; d < DDIM; d += 8) { float v = xp[d]; s += v * v; }
        s += __shfl_xor(s, 4, 8);
        s += __shfl_xor(s, 2, 8);
        s += __shfl_xor(s, 1, 8);
        if ((lane & 7) == 0) s_norm2[r] = s;
    }

    // Per-lane quadratic-form partials: part[mt][v] accumulates
    // sum over owned N of xr[row, n] * x[row, n], row = mt*16 + v + 8*hi.
    float part[MT][8];
    #pragma unroll
    for (int mt = 0; mt < MT; ++mt)
        #pragma unroll
        for (int v = 0; v < 8; ++v) part[mt][v] = 0.f;

    const int koff = hi << 1;   // A/B fragment K offset for this half-wave

    for (int pass = 0; pass < NPASS; ++pass) {
        // N columns owned by this wave this pass (tile-local col = l15)
        int ncol[NT];
        #pragma unroll
        for (int t = 0; t < NT; ++t)
            ncol[t] = (((pass * NWAVES + wave) * NT + t) << 4) + l15;

        v8f c[MT][NT];
        #pragma unroll
        for (int mt = 0; mt < MT; ++mt)
            #pragma unroll
            for (int t = 0; t < NT; ++t)
                c[mt][t] = (v8f){0.f, 0.f, 0.f, 0.f, 0.f, 0.f, 0.f, 0.f};

        // --------- K loop: 512 steps, 8 independent WMMAs per step ---------
        #pragma unroll 2
        for (int kk = 0; kk < DDIM; kk += 4) {
            const int kb = kk + koff;
            // A fragments (16x4 f32): lane holds row m0+mt*16+l15,
            // K = kb, kb+1 -> contiguous float2 (global_load_b64)
            v2f a[MT];
            #pragma unroll
            for (int mt = 0; mt < MT; ++mt)
                a[mt] = *(const v2f*)(X + (size_t)(m0 + mt * 16 + l15) * DDIM + kb);

            #pragma unroll
            for (int t = 0; t < NT; ++t) {
                // B fragment (4x16 f32): rows kb, kb+1 of rho at column ncol[t]
                v2f b;
                b.x = R[(size_t)kb * DDIM + ncol[t]];
                b.y = R[(size_t)(kb + 1) * DDIM + ncol[t]];
                #pragma unroll
                for (int mt = 0; mt < MT; ++mt)
                    c[mt][t] = __builtin_amdgcn_wmma_f32_16x16x4_f32(
                        false, a[mt], false, b, (short)0, c[mt][t], false, false);
            }
        }

        // --------- epilogue: fold xr tile into quad partials ---------------
        // C layout: VGPR v, lane -> M = v + 8*hi, N = lane&15 (== ncol[t] col)
        #pragma unroll
        for (int mt = 0; mt < MT; ++mt)
            #pragma unroll
            for (int t = 0; t < NT; ++t)
                #pragma unroll
                for (int v = 0; v < 8; ++v) {
                    const int row = m0 + mt * 16 + (hi << 3) + v;
                    part[mt][v] += c[mt][t][v] * X[(size_t)row * DDIM + ncol[t]];
                }
    }

    // ---------------- reduce partials across each 16-lane half -------------
    #pragma unroll
    for (int mt = 0; mt < MT; ++mt)
        #pragma unroll
        for (int v = 0; v < 8; ++v) {
            float s = part[mt][v];
            s += __shfl_xor(s, 1, 16);
            s += __shfl_xor(s, 2, 16);
            s += __shfl_xor(s, 4, 16);
            s += __shfl_xor(s, 8, 16);
            if (l15 == 0) s_quad[wave][mt * 16 + (hi << 3) + v] = s;
        }

    __syncthreads();

    // ---------------- combine waves, apply norm2 ---------------------------
    if (tid < MB) {
        float q = 0.f;
        #pragma unroll
        for (int w = 0; w < NWAVES; ++w) q += s_quad[w][tid];
        out[m0 + tid] = s_norm2[tid] * q;
    }
}

extern "C" void kernel_launch(void* const* d_in, const int* in_sizes, int n_in,
                              void* d_out, int out_size, void* d_ws, size_t ws_size,
                              hipStream_t stream) {
    (void)in_sizes; (void)n_in; (void)out_size; (void)d_ws; (void)ws_size;
    const float* X = (const float*)d_in[0];   // inputs (16384, 2048) f32
    const float* R = (const float*)d_in[1];   // rho    (2048, 2048) f32
    float* out = (float*)d_out;               // (16384,) f32

    QMeasureDensity_wmma_kernel<<<B_ROWS / MB, 256, 0, stream>>>(X, R, out);
}